// GCN_32650341384680
// MI455X (gfx1250) — compile-verified
//
#include <hip/hip_runtime.h>
#include <hip/hip_bf16.h>
#include <math.h>

// ---------------------------------------------------------------------------
// GCN on MI455X (gfx1250, wave32).
// Dense GEMMs: V_WMMA_F32_16X16X4_F32, one 32x16 output tile per wave
// (two 16x16 WMMA subtiles sharing the B fragment). All GEMM shapes are
// compile-time template parameters so every load/store address is a base
// pointer + 24-bit immediate offset (no per-iteration address math).
// Sparse aggregation: L2-resident GLOBAL_ATOMIC_ADD_F32 (all per-layer
// feature tables fit in the 192MB L2).
// ---------------------------------------------------------------------------

typedef __attribute__((ext_vector_type(2))) float v2f;
typedef __attribute__((ext_vector_type(8))) float v8f;

static __host__ __device__ inline int cdiv(int a, int b) { return (a + b - 1) / b; }

// deg[i] = 1.0 (self-loop contribution)
__global__ void k_init_deg(float* __restrict__ deg, int n) {
    int i = blockIdx.x * blockDim.x + threadIdx.x;
    if (i < n) deg[i] = 1.0f;
}

// deg[dst[e]] += 1
__global__ void k_deg_scatter(const int* __restrict__ dst, float* __restrict__ deg, int E) {
    int e = blockIdx.x * blockDim.x + threadIdx.x;
    if (e < E) atomicAdd(&deg[dst[e]], 1.0f);
}

// in-place: dinv[i] = rsqrt(deg[i])   (deg >= 1 always due to self-loops)
__global__ void k_deg_to_dinv(float* __restrict__ d, int n) {
    int i = blockIdx.x * blockDim.x + threadIdx.x;
    if (i < n) d[i] = rsqrtf(d[i]);
}

// zero-pad W[K,N] -> Wp[K,Npad] so GEMM loads need no per-lane guards
__global__ void k_pad_w(const float* __restrict__ W, float* __restrict__ Wp,
                        int K, int N, int Npad) {
    int idx = blockIdx.x * blockDim.x + threadIdx.x;
    if (idx < K * Npad) {
        int r = idx / Npad, c = idx % Npad;
        Wp[idx] = (c < N) ? W[(size_t)r * N + c] : 0.0f;
    }
}

// ---------------------------------------------------------------------------
// WMMA f32 GEMM:  Out[M,N] = A[M,K] @ W[K,NPAD]  (+ optional bias[N])
// K, N, NPAD are compile-time -> fully unrolled K-loop, all offsets are
// instruction immediates. Store guard (col < N) only in the epilogue.
//
// Fragment layouts (ISA 7.12.2, wave32):
//   A 16x4 : lanes 0-15 -> M=lane, vgpr{0,1}=K{0,1}; lanes 16-31 -> K{2,3}
//   B 4x16 : lanes 0-15 -> N=lane, vgpr{0,1}=K{0,1}; lanes 16-31 -> K{2,3}
//   C 16x16: lanes 0-15 -> N=lane, vgpr r -> M=r; lanes 16-31 -> M=r+8
// Tile guard is wave-uniform so EXEC is all-1s at every v_wmma.
// ---------------------------------------------------------------------------
template <int K, int N, int NPAD>
__global__ void k_gemm_wmma_f32(const float* __restrict__ A,
                                const float* __restrict__ W,    // [K, NPAD]
                                const float* __restrict__ bias, // may be null
                                float* __restrict__ Out,        // [M, N]
                                int M) {
    const int lane = threadIdx.x & 31;
    const int wave = threadIdx.x >> 5;
    const int wpb  = blockDim.x >> 5;
    constexpr int tilesN = NPAD >> 4;
    const int tilesM = M >> 5;                 // 32-row tiles (M % 32 == 0 here)
    const int tile = blockIdx.x * wpb + wave;  // wave-uniform
    if (tile >= tilesM * tilesN) return;       // wave-uniform exit: EXEC stays full

    const int tm = tile / tilesN;
    const int tn = tile % tilesN;
    const int m0 = tm << 5;
    const int n0 = tn << 4;

    const int half = lane >> 4;      // 0: lanes 0-15, 1: lanes 16-31
    const int l15  = lane & 15;

    // half*2 folded into base pointers -> per-k offsets are immediates
    const float* __restrict__ a0p = A + (size_t)(m0 + l15) * K + half * 2;
    const float* __restrict__ a1p = a0p + (size_t)16 * K;
    const float* __restrict__ wp  = W + (size_t)(half * 2) * NPAD + n0 + l15;

    v8f acc0 = {};
    v8f acc1 = {};
#pragma unroll
    for (int k = 0; k < K; k += 4) {
        v2f af0, af1, bf;
        af0.x = a0p[k];
        af0.y = a0p[k + 1];
        af1.x = a1p[k];
        af1.y = a1p[k + 1];
        bf.x  = wp[k * NPAD];
        bf.y  = wp[(k + 1) * NPAD];
        acc0 = __builtin_amdgcn_wmma_f32_16x16x4_f32(false, af0, false, bf,
                                                     (short)0, acc0, false, false);
        acc1 = __builtin_amdgcn_wmma_f32_16x16x4_f32(false, af1, false, bf,
                                                     (short)0, acc1, false, false);
    }

    const int bcol = n0 + l15;
    if (bcol < N) {
        const float bb = bias ? bias[bcol] : 0.0f;
        float* __restrict__ o0 = Out + (size_t)(m0 + half * 8) * N + bcol;
        float* __restrict__ o1 = o0 + 16 * N;
#pragma unroll
        for (int r = 0; r < 8; ++r) {
            o0[r * N] = acc0[r] + bb;
            o1[r * N] = acc1[r] + bb;
        }
    }
}

// a[i,c] = t[i,c] * dinv[i]^2   (self-loop term; doubles as zero-init)
template <int FSHIFT>
__global__ void k_self_init(const float* __restrict__ t,
                            const float* __restrict__ dinv,
                            float* __restrict__ a, int n) {
    int idx = blockIdx.x * blockDim.x + threadIdx.x;
    if (idx < (n << FSHIFT)) {
        float d = dinv[idx >> FSHIFT];
        a[idx] = t[idx] * d * d;
    }
}

// one wave per edge: a[dst,c] += t[src,c] * dinv[src]*dinv[dst]
// indices are wave-uniform -> readfirstlane to force SGPR/scalar path;
// lanes stride the feature dim -> coalesced 128B gathers, L2-resident atomics
template <int F>
__global__ void k_edge_scatter(const float* __restrict__ t,
                               const float* __restrict__ dinv,
                               const int* __restrict__ src,
                               const int* __restrict__ dst,
                               float* __restrict__ a, int E) {
    const int lane = threadIdx.x & 31;
    const int e = (blockIdx.x * blockDim.x + threadIdx.x) >> 5;  // wave-uniform
    if (e >= E) return;
    const int s = __builtin_amdgcn_readfirstlane(src[e]);
    const int d = __builtin_amdgcn_readfirstlane(dst[e]);
    const float w = dinv[s] * dinv[d];
    const float* __restrict__ ts = t + (size_t)s * F + lane;
    float* __restrict__ ad = a + (size_t)d * F + lane;
#pragma unroll
    for (int i = 0; i < F / 32; ++i)
        atomicAdd(&ad[i * 32], ts[i * 32] * w);
}

// h[i,c] = tanh(a[i,c] + b[c]);  F power of two
template <int F>
__global__ void k_bias_tanh(const float* __restrict__ a,
                            const float* __restrict__ b,
                            float* __restrict__ h, int n) {
    int idx = blockIdx.x * blockDim.x + threadIdx.x;
    if (idx < n * F) h[idx] = tanhf(a[idx] + b[idx & (F - 1)]);
}

extern "C" void kernel_launch(void* const* d_in, const int* in_sizes, int n_in,
                              void* d_out, int out_size, void* d_ws, size_t ws_size,
                              hipStream_t stream) {
    const float* x  = (const float*)d_in[0];
    const int*   ei = (const int*)d_in[1];   // [2,E] int32 (JAX x64 disabled)
    const float* W1 = (const float*)d_in[2];
    const float* b1 = (const float*)d_in[3];
    const float* W2 = (const float*)d_in[4];
    const float* b2 = (const float*)d_in[5];
    const float* W3 = (const float*)d_in[6];
    const float* b3 = (const float*)d_in[7];
    const float* Wc = (const float*)d_in[8];
    const float* bc = (const float*)d_in[9];

    const int n = in_sizes[0] / 256;  // 100000
    const int E = in_sizes[1] / 2;    // 1600000
    const int* src = ei;
    const int* dst = ei + E;

    // workspace (floats): dinv(n) | t(128n) | a(128n) | h1(128n) | h2(64n) | Wcp(32*96)
    float* ws   = (float*)d_ws;
    float* dinv = ws;
    float* t    = dinv + (size_t)n;
    float* a    = t    + (size_t)n * 128;
    float* h1   = a    + (size_t)n * 128;
    float* h2   = h1   + (size_t)n * 128;
    float* Wcp  = h2   + (size_t)n * 64;

    float* out = (float*)d_out;            // [n, 86]
    float* h3  = out + (size_t)n * 86;     // [n, 32] (second tuple element)

    const int T   = 256;                   // 8 waves per block
    const int wpb = T >> 5;

    // symmetric normalization
    k_init_deg   <<<cdiv(n, T), T, 0, stream>>>(dinv, n);
    k_deg_scatter<<<cdiv(E, T), T, 0, stream>>>(dst, dinv, E);
    k_deg_to_dinv<<<cdiv(n, T), T, 0, stream>>>(dinv, n);

    // zero-pad classifier weight 32x86 -> 32x96
    k_pad_w<<<cdiv(32 * 96, T), T, 0, stream>>>(Wc, Wcp, 32, 86, 96);

    const int tilesM = n >> 5;             // 3125 (32-row tiles)

    // ---- layer 1: 256 -> 128 ----
    k_gemm_wmma_f32<256, 128, 128><<<cdiv(tilesM * 8, wpb), T, 0, stream>>>(x, W1, nullptr, t, n);
    k_self_init<7>    <<<cdiv(n * 128, T), T, 0, stream>>>(t, dinv, a, n);
    k_edge_scatter<128><<<cdiv(E, wpb), T, 0, stream>>>(t, dinv, src, dst, a, E);
    k_bias_tanh<128>  <<<cdiv(n * 128, T), T, 0, stream>>>(a, b1, h1, n);

    // ---- layer 2: 128 -> 64 ----
    k_gemm_wmma_f32<128, 64, 64><<<cdiv(tilesM * 4, wpb), T, 0, stream>>>(h1, W2, nullptr, t, n);
    k_self_init<6>    <<<cdiv(n * 64, T), T, 0, stream>>>(t, dinv, a, n);
    k_edge_scatter<64><<<cdiv(E, wpb), T, 0, stream>>>(t, dinv, src, dst, a, E);
    k_bias_tanh<64>   <<<cdiv(n * 64, T), T, 0, stream>>>(a, b2, h2, n);

    // ---- layer 3: 64 -> 32 ----
    k_gemm_wmma_f32<64, 32, 32><<<cdiv(tilesM * 2, wpb), T, 0, stream>>>(h2, W3, nullptr, t, n);
    k_self_init<5>    <<<cdiv(n * 32, T), T, 0, stream>>>(t, dinv, a, n);
    k_edge_scatter<32><<<cdiv(E, wpb), T, 0, stream>>>(t, dinv, src, dst, a, E);
    k_bias_tanh<32>   <<<cdiv(n * 32, T), T, 0, stream>>>(a, b3, h3, n);

    // ---- classifier: 32 -> 86 (bias fused, padded weight) ----
    k_gemm_wmma_f32<32, 86, 96><<<cdiv(tilesM * 6, wpb), T, 0, stream>>>(h3, Wcp, bc, out, n);
}